// CoordinateLoss_75282186764793
// MI455X (gfx1250) — compile-verified
//
#include <hip/hip_runtime.h>

typedef __attribute__((ext_vector_type(2))) float v2f;
typedef __attribute__((ext_vector_type(8))) float v8f;

constexpr int BATCH = 256;
constexpr int SEQ   = 16384;
constexpr float DELTA = 1.0f; // Huber delta

// ws layout (floats):
//   stats[BATCH][16] : cnt, Sp[3], St[3], Spt[9]   @ 0
//   rt[BATCH][12]    : R[9] row-major, t[3]        @ BATCH*16
//   lossPartial[BATCH]                             @ BATCH*16 + BATCH*12
constexpr int WS_STATS = 0;
constexpr int WS_RT    = BATCH * 16;
constexpr int WS_LOSS  = BATCH * 16 + BATCH * 12;

constexpr int ROW = 100;   // LDS row stride in floats: 96 data + 4 zero pad
constexpr int PAD = 96;    // index of guaranteed-zero word in each row

__device__ __forceinline__ float waveReduceAdd(float v) {
#pragma unroll
  for (int off = 16; off > 0; off >>= 1) v += __shfl_down(v, off, 32);
  return v;
}

// ---------------------------------------------------------------------------
// Phase 1: per-batch masked stats. Covariance cross-products on the FP32
// matrix pipe (V_WMMA_F32_16X16X4_F32, 4 points per instruction). One block
// per batch, 8 waves; each wave stages 32 coalesced points. Target tile is
// staged memory->LDS via GLOBAL_LOAD_ASYNC_TO_LDS_B32 (ASYNCcnt), pred tile
// via VGPRs (needs the mask multiply anyway). Wave-synchronous inner loop:
// no block barriers (DS ops from one wave are in-order; async ordered with
// explicit s_wait_asynccnt / s_wait_dscnt).
// ---------------------------------------------------------------------------
__global__ __launch_bounds__(256) void k_stats(const float* __restrict__ pred,
                                               const float* __restrict__ targ,
                                               const int*   __restrict__ mask,
                                               float*       __restrict__ ws) {
  const int b    = blockIdx.x;
  const int tid  = threadIdx.x;
  const int wave = tid >> 5;
  const int lane = tid & 31;

  __shared__ float pmLDS[8][ROW];  // masked pred, 32 pts x 3 (+ zero pad)
  __shared__ float ttLDS[8][ROW];  // raw target, 32 pts x 3 (+ zero pad)
  __shared__ float sptLDS[8][9];
  __shared__ float wstat[8][7];

  float* pmW = &pmLDS[wave][0];
  float* ttW = &ttLDS[wave][0];
  // LDS byte offset of this wave's target row (low 32 bits of the generic pointer)
  const unsigned ttBase = (unsigned)(uintptr_t)(void*)ttW;
  const unsigned long long tBatch =
      (unsigned long long)(uintptr_t)(targ + (size_t)b * SEQ * 3);

  // zero the pad words (wave-private, wave-synchronous -> no barrier needed)
  if (lane < 4) { pmW[PAD + lane] = 0.f; ttW[PAD + lane] = 0.f; }

  const size_t baseP = (size_t)b * SEQ * 3;
  const size_t baseM = (size_t)b * SEQ;

  v8f acc = {0.f, 0.f, 0.f, 0.f, 0.f, 0.f, 0.f, 0.f};
  float cnt = 0.f, sp0 = 0.f, sp1 = 0.f, sp2 = 0.f, st0 = 0.f, st1 = 0.f, st2 = 0.f;

  const int half = lane >> 4;   // 0: K=0,1  1: K=2,3  (A/B VGPR lane split)
  const int r    = lane & 15;   // row (A) / col (B) index within 16x16 tile

  for (int it = 0; it < SEQ / 256; ++it) {
    const int s = it * 256 + wave * 32 + lane;   // this lane stages point s
    const float m  = (float)mask[baseM + s];
    const size_t o = baseP + (size_t)s * 3;
    const float p0 = pred[o + 0], p1 = pred[o + 1], p2 = pred[o + 2];
    __builtin_prefetch((const char*)(pred + o) + 24576, 0, 0);

    const float pm0 = p0 * m, pm1 = p1 * m, pm2 = p2 * m;
    cnt += m;
    sp0 += pm0; sp1 += pm1; sp2 += pm2;

    // ensure prior iteration's DS reads of ttW are done before async overwrite
    asm volatile("s_wait_dscnt 0x0" ::: "memory");

    // async-copy this wave's 32-point target tile (96 dwords) into LDS
    {
      const unsigned chunk = (unsigned)((it * 256 + wave * 32) * 12);
      const unsigned l0 = ttBase + (unsigned)(lane * 4);
      const unsigned g0 = chunk + (unsigned)(lane * 4);
      asm volatile(
          "global_load_async_to_lds_b32 %0, %1, %6\n\t"
          "global_load_async_to_lds_b32 %2, %3, %6\n\t"
          "global_load_async_to_lds_b32 %4, %5, %6"
          :
          : "v"(l0), "v"(g0),
            "v"(l0 + 128u), "v"(g0 + 128u),
            "v"(l0 + 256u), "v"(g0 + 256u),
            "s"(tBatch)
          : "memory");
    }

    // stage masked pred through DS (same-wave DS ops are in-order)
    pmW[lane * 3 + 0] = pm0;
    pmW[lane * 3 + 1] = pm1;
    pmW[lane * 3 + 2] = pm2;

    // target tile must be resident before DS reads below
    asm volatile("s_wait_asynccnt 0x0" ::: "memory");

    // masked target sums (read back from LDS)
    const float t0 = ttW[lane * 3 + 0];
    const float t1 = ttW[lane * 3 + 1];
    const float t2 = ttW[lane * 3 + 2];
    st0 += t0 * m; st1 += t1 * m; st2 += t2 * m;

    // A[mrow][k] = masked pred coord mrow of point k (mrow<3), else 0
    // B[k][ncol] = target coord ncol of point k (ncol<3), else 0
    // 16x16x4 f32 layout: lanes 0-15 -> K=0,1 ; lanes 16-31 -> K=2,3.
    // Out-of-tile lanes read the zeroed pad word (address select, no branch).
#pragma unroll
    for (int q = 0; q < 8; ++q) {
      const int k0  = q * 4 + half * 2;
      const int iax = (r < 3) ? ((k0 + 0) * 3 + r) : PAD;
      const int iay = (r < 3) ? ((k0 + 1) * 3 + r) : PAD;
      v2f a, bb;
      a.x  = pmW[iax];
      a.y  = pmW[iay];
      bb.x = ttW[iax];
      bb.y = ttW[iay];
      acc = __builtin_amdgcn_wmma_f32_16x16x4_f32(false, a, false, bb,
                                                  (short)0, acc, false, false);
    }
  }

  // C/D layout: VGPR i = row M=i (lanes 0-15), lane = col N. 3x3 lives in
  // acc[0..2] of lanes 0..2.
  if (lane < 3) {
    sptLDS[wave][0 * 3 + lane] = acc[0];
    sptLDS[wave][1 * 3 + lane] = acc[1];
    sptLDS[wave][2 * 3 + lane] = acc[2];
  }
  cnt = waveReduceAdd(cnt);
  sp0 = waveReduceAdd(sp0); sp1 = waveReduceAdd(sp1); sp2 = waveReduceAdd(sp2);
  st0 = waveReduceAdd(st0); st1 = waveReduceAdd(st1); st2 = waveReduceAdd(st2);
  if (lane == 0) {
    wstat[wave][0] = cnt;
    wstat[wave][1] = sp0; wstat[wave][2] = sp1; wstat[wave][3] = sp2;
    wstat[wave][4] = st0; wstat[wave][5] = st1; wstat[wave][6] = st2;
  }
  __syncthreads();

  if (tid == 0) {
    float o[16];
#pragma unroll
    for (int i = 0; i < 16; ++i) o[i] = 0.f;
    for (int w = 0; w < 8; ++w) {      // fixed order -> deterministic
#pragma unroll
      for (int i = 0; i < 7; ++i) o[i] += wstat[w][i];
#pragma unroll
      for (int e = 0; e < 9; ++e) o[7 + e] += sptLDS[w][e];
    }
    float* st = ws + WS_STATS + b * 16;
#pragma unroll
    for (int i = 0; i < 16; ++i) st[i] = o[i];
  }
}

// ---------------------------------------------------------------------------
// Phase 2: one thread per batch: H, Jacobi SVD(3x3), Kabsch R (det fix), t.
// ---------------------------------------------------------------------------
__device__ __forceinline__ float det3(const float M[3][3]) {
  return M[0][0] * (M[1][1] * M[2][2] - M[1][2] * M[2][1])
       - M[0][1] * (M[1][0] * M[2][2] - M[1][2] * M[2][0])
       + M[0][2] * (M[1][0] * M[2][1] - M[1][1] * M[2][0]);
}

__global__ __launch_bounds__(256) void k_svd(const float* __restrict__ ws_stats,
                                             float* __restrict__ ws_rt) {
  const int b = threadIdx.x;
  const float* st = ws_stats + b * 16;
  const float cnt = st[0];
  const float inv = 1.0f / cnt;
  float Sp[3] = {st[1], st[2], st[3]};
  float St[3] = {st[4], st[5], st[6]};
  float H[3][3];
#pragma unroll
  for (int i = 0; i < 3; ++i)
#pragma unroll
    for (int j = 0; j < 3; ++j)
      H[i][j] = st[7 + i * 3 + j] - Sp[i] * St[j] * inv;

  // A = H^T H (symmetric PSD)
  float A[3][3];
#pragma unroll
  for (int i = 0; i < 3; ++i)
#pragma unroll
    for (int j = 0; j < 3; ++j)
      A[i][j] = H[0][i] * H[0][j] + H[1][i] * H[1][j] + H[2][i] * H[2][j];

  float V[3][3] = {{1, 0, 0}, {0, 1, 0}, {0, 0, 1}};
  const int JP[3] = {0, 0, 1}, JQ[3] = {1, 2, 2};
  for (int sweep = 0; sweep < 12; ++sweep) {
    for (int rot = 0; rot < 3; ++rot) {
      const int p = JP[rot], q = JQ[rot];
      const float apq = A[p][q];
      if (fabsf(apq) > 1e-20f) {
        const float tau = (A[q][q] - A[p][p]) / (2.0f * apq);
        const float tt  = (tau >= 0.f ? 1.f : -1.f) / (fabsf(tau) + sqrtf(1.f + tau * tau));
        const float c   = 1.0f / sqrtf(1.f + tt * tt);
        const float s   = tt * c;
#pragma unroll
        for (int k = 0; k < 3; ++k) {
          const float akp = A[k][p], akq = A[k][q];
          A[k][p] = c * akp - s * akq;
          A[k][q] = s * akp + c * akq;
        }
#pragma unroll
        for (int k = 0; k < 3; ++k) {
          const float apk = A[p][k], aqk = A[q][k];
          A[p][k] = c * apk - s * aqk;
          A[q][k] = s * apk + c * aqk;
        }
#pragma unroll
        for (int k = 0; k < 3; ++k) {
          const float vkp = V[k][p], vkq = V[k][q];
          V[k][p] = c * vkp - s * vkq;
          V[k][q] = s * vkp + c * vkq;
        }
      }
    }
  }

  // sort eigenvalues descending, permute V columns
  float ev[3] = {A[0][0], A[1][1], A[2][2]};
  int idx[3] = {0, 1, 2};
  for (int i = 0; i < 2; ++i)
    for (int j = 0; j < 2 - i; ++j)
      if (ev[idx[j]] < ev[idx[j + 1]]) { int tswp = idx[j]; idx[j] = idx[j + 1]; idx[j + 1] = tswp; }

  float Vc[3][3];
#pragma unroll
  for (int j = 0; j < 3; ++j)
#pragma unroll
    for (int i = 0; i < 3; ++i) Vc[i][j] = V[i][idx[j]];

  // U columns = H * Vc / sigma  (fallback if degenerate)
  float U[3][3];
#pragma unroll
  for (int j = 0; j < 3; ++j) {
    float u0 = H[0][0] * Vc[0][j] + H[0][1] * Vc[1][j] + H[0][2] * Vc[2][j];
    float u1 = H[1][0] * Vc[0][j] + H[1][1] * Vc[1][j] + H[1][2] * Vc[2][j];
    float u2 = H[2][0] * Vc[0][j] + H[2][1] * Vc[1][j] + H[2][2] * Vc[2][j];
    const float n = sqrtf(u0 * u0 + u1 * u1 + u2 * u2);
    if (n > 1e-12f) {
      const float rn = 1.0f / n;
      U[0][j] = u0 * rn; U[1][j] = u1 * rn; U[2][j] = u2 * rn;
    } else if (j == 2) { // degenerate: complete basis
      U[0][2] = U[1][0] * U[2][1] - U[2][0] * U[1][1];
      U[1][2] = U[2][0] * U[0][1] - U[0][0] * U[2][1];
      U[2][2] = U[0][0] * U[1][1] - U[1][0] * U[0][1];
    } else {
      U[0][j] = (j == 0) ? 1.f : 0.f; U[1][j] = (j == 1) ? 1.f : 0.f; U[2][j] = 0.f;
    }
  }

  // reflection fix: flip last column of V if det(V U^T) < 0
  if (det3(U) * det3(Vc) < 0.f) { Vc[0][2] = -Vc[0][2]; Vc[1][2] = -Vc[1][2]; Vc[2][2] = -Vc[2][2]; }

  // R = V * U^T
  float R[3][3];
#pragma unroll
  for (int i = 0; i < 3; ++i)
#pragma unroll
    for (int k = 0; k < 3; ++k)
      R[i][k] = Vc[i][0] * U[k][0] + Vc[i][1] * U[k][1] + Vc[i][2] * U[k][2];

  float cp[3], ct[3];
#pragma unroll
  for (int i = 0; i < 3; ++i) { cp[i] = Sp[i] * inv; ct[i] = St[i] * inv; }

  float* rt = ws_rt + b * 12;
#pragma unroll
  for (int i = 0; i < 3; ++i) {
#pragma unroll
    for (int j = 0; j < 3; ++j) rt[i * 3 + j] = R[i][j];
    rt[9 + i] = ct[i] - (R[i][0] * cp[0] + R[i][1] * cp[1] + R[i][2] * cp[2]);
  }
}

// ---------------------------------------------------------------------------
// Phase 3: streaming masked Huber loss of (R*p + t) vs target. 1 block/batch.
// ---------------------------------------------------------------------------
__global__ __launch_bounds__(256) void k_loss(const float* __restrict__ pred,
                                              const float* __restrict__ targ,
                                              const int*   __restrict__ mask,
                                              const float* __restrict__ ws_rt,
                                              float*       __restrict__ ws_loss) {
  const int b    = blockIdx.x;
  const int tid  = threadIdx.x;
  const int wave = tid >> 5;
  const int lane = tid & 31;
  __shared__ float wred[8];

  float R[9], T[3];
  {
    const float* rt = ws_rt + b * 12;
#pragma unroll
    for (int i = 0; i < 9; ++i) R[i] = rt[i];
#pragma unroll
    for (int i = 0; i < 3; ++i) T[i] = rt[9 + i];
  }

  const size_t baseP = (size_t)b * SEQ * 3;
  const size_t baseM = (size_t)b * SEQ;

  float acc = 0.f;
  for (int it = 0; it < SEQ / 256; ++it) {
    const int s = it * 256 + tid;
    const float m  = (float)mask[baseM + s];
    const size_t o = baseP + (size_t)s * 3;
    __builtin_prefetch((const char*)(pred + o) + 24576, 0, 0);
    __builtin_prefetch((const char*)(targ + o) + 24576, 0, 0);
    const float p0 = pred[o + 0], p1 = pred[o + 1], p2 = pred[o + 2];
    const float t0 = targ[o + 0], t1 = targ[o + 1], t2 = targ[o + 2];
    const float a0 = R[0] * p0 + R[1] * p1 + R[2] * p2 + T[0];
    const float a1 = R[3] * p0 + R[4] * p1 + R[5] * p2 + T[1];
    const float a2 = R[6] * p0 + R[7] * p1 + R[8] * p2 + T[2];
    float h = 0.f;
    {
      const float d = a0 - t0, ad = fabsf(d);
      h += (ad < DELTA) ? 0.5f * d * d : DELTA * (ad - 0.5f * DELTA);
    }
    {
      const float d = a1 - t1, ad = fabsf(d);
      h += (ad < DELTA) ? 0.5f * d * d : DELTA * (ad - 0.5f * DELTA);
    }
    {
      const float d = a2 - t2, ad = fabsf(d);
      h += (ad < DELTA) ? 0.5f * d * d : DELTA * (ad - 0.5f * DELTA);
    }
    acc += m * h;
  }

  acc = waveReduceAdd(acc);
  if (lane == 0) wred[wave] = acc;
  __syncthreads();
  if (tid == 0) {
    float s = 0.f;
    for (int w = 0; w < 8; ++w) s += wred[w];  // fixed order
    ws_loss[b] = s;
  }
}

// ---------------------------------------------------------------------------
// Phase 4: final fixed-order reduction over 256 batches; out = loss / cnt.
// ---------------------------------------------------------------------------
__global__ __launch_bounds__(256) void k_final(const float* __restrict__ ws_stats,
                                               const float* __restrict__ ws_loss,
                                               float* __restrict__ out) {
  const int tid  = threadIdx.x;
  const int wave = tid >> 5;
  const int lane = tid & 31;
  __shared__ float l8[8], c8[8];
  float l = ws_loss[tid];
  float c = ws_stats[tid * 16];
  l = waveReduceAdd(l);
  c = waveReduceAdd(c);
  if (lane == 0) { l8[wave] = l; c8[wave] = c; }
  __syncthreads();
  if (tid == 0) {
    float L = 0.f, C = 0.f;
    for (int w = 0; w < 8; ++w) { L += l8[w]; C += c8[w]; }
    out[0] = L / C;
  }
}

extern "C" void kernel_launch(void* const* d_in, const int* in_sizes, int n_in,
                              void* d_out, int out_size, void* d_ws, size_t ws_size,
                              hipStream_t stream) {
  const float* pred = (const float*)d_in[0];
  const float* targ = (const float*)d_in[1];
  const int*   mask = (const int*)d_in[2];
  float* out = (float*)d_out;
  float* ws  = (float*)d_ws;

  k_stats<<<dim3(BATCH), dim3(256), 0, stream>>>(pred, targ, mask, ws);
  k_svd<<<dim3(1), dim3(BATCH), 0, stream>>>(ws + WS_STATS, ws + WS_RT);
  k_loss<<<dim3(BATCH), dim3(256), 0, stream>>>(pred, targ, mask, ws + WS_RT, ws + WS_LOSS);
  k_final<<<dim3(1), dim3(256), 0, stream>>>(ws + WS_STATS, ws + WS_LOSS, out);
}